// GCN_4_4_8_8_16_16_32_72782515798130
// MI455X (gfx1250) — compile-verified
//
#include <hip/hip_runtime.h>
#include <hip/hip_bf16.h>

typedef __attribute__((ext_vector_type(2))) float v2f;
typedef __attribute__((ext_vector_type(8))) float v8f;

#define N_NODES 24
#define BT      16    // batch elements per block (one wave)
#define BLOCK   32    // one wave32 per block
#define APITCH  25    // adjacency / batch-row pitch (gcd(25,64)=1 -> conflict-free)
#define HPITCH  400   // h feature pitch: f*400 + b*25 + i
#define WPITCH  401   // hw feature pitch: f*401 + b*25 + j  (401 mod 64 = 17)
#define FCPITCH 129   // fc1-out row pitch (129 mod 64 = 1)

static __device__ __forceinline__ v8f wmma4(v2f a, v2f b, v8f c) {
    // D = A(16x4 f32) * B(4x16 f32) + C(16x16 f32), exact fp32 semantics
    return __builtin_amdgcn_wmma_f32_16x16x4_f32(
        /*neg_a=*/false, a, /*neg_b=*/false, b,
        /*c_mod=*/(short)0, c, /*reuse_a=*/false, /*reuse_b=*/false);
}

// ---------------------------------------------------------------------------
// Kernel 1: normalized GCN adjacency (self-loops, duplicate-edge accumulation),
// stored zero-padded to 32 rows with pitch 25: Apad[i*25+j] = norm(i<-j).
// ---------------------------------------------------------------------------
__global__ void build_adj_kernel(const int* __restrict__ ei, float* __restrict__ Apad) {
    __shared__ float deg[N_NODES];
    __shared__ float dinv[N_NODES];
    __shared__ float Ash[32 * APITCH];
    const int t = threadIdx.x;  // 128 threads
    if (t < N_NODES) deg[t] = 1.0f;  // self-loop
    for (int i = t; i < 32 * APITCH; i += 128) Ash[i] = 0.0f;
    __syncthreads();
    if (t < 96) atomicAdd(&deg[ei[96 + t]], 1.0f);  // dst degree
    __syncthreads();
    if (t < N_NODES) dinv[t] = rsqrtf(deg[t]);
    __syncthreads();
    if (t < 96) {
        int s = ei[t], d = ei[96 + t];
        atomicAdd(&Ash[d * APITCH + s], dinv[s] * dinv[d]);
    }
    if (t < N_NODES) atomicAdd(&Ash[t * APITCH + t], dinv[t] * dinv[t]);
    __syncthreads();
    for (int i = t; i < 32 * APITCH; i += 128) Apad[i] = Ash[i];
}

// ---------------------------------------------------------------------------
// Kernel 2: fused GCN (7 layers) + fc1 (WMMA) + fc2 + log_softmax.
// One wave / 16 batch elements per block; all activations LDS-resident in
// bank-conflict-free feature-major layout. Row->(b,i) maps kept incrementally
// (no integer divisions in hot loops).
// ---------------------------------------------------------------------------
__global__ void gcn_fused_kernel(
    const float* __restrict__ x, const float* __restrict__ Apad_g,
    const float* __restrict__ W1, const float* __restrict__ b1,
    const float* __restrict__ W2, const float* __restrict__ b2,
    const float* __restrict__ W3, const float* __restrict__ b3,
    const float* __restrict__ W4, const float* __restrict__ b4,
    const float* __restrict__ W5, const float* __restrict__ b5,
    const float* __restrict__ W6, const float* __restrict__ b6,
    const float* __restrict__ W7, const float* __restrict__ b7,
    const float* __restrict__ fcw1, const float* __restrict__ fcb1,
    const float* __restrict__ fcw2, const float* __restrict__ fcb2,
    float* __restrict__ out)
{
    extern __shared__ float smem[];
    float* sA   = smem;                  // 32*25  = 800   padded adjacency
    float* sW   = smem + 800;            // 32*32  = 1024  zero-padded weight tile
    float* sBia = smem + 1824;           // 32             layer bias
    float* hbuf = smem + 1856;           // 32*400 = 12800 h, feature-major
    float* wbuf = smem + 14656;          // 32*401 = 12832 h@W staging
    float* sFc  = wbuf;                  // 16*129 = 2064  fc1 out (overlays dead wbuf)
    // total 27488 floats = 109952 bytes

    const int t    = threadIdx.x;        // 0..31, single wave
    const int n16  = t & 15;             // lane % 16
    const int hi   = t >> 4;             // lane group (0/1)
    const long b0  = (long)blockIdx.x * BT;

    const int   FIN[7]  = {1, 4, 4, 8, 8, 16, 16};
    const int   FOUT[7] = {4, 4, 8, 8, 16, 16, 32};
    const float* Ws[7]  = {W1, W2, W3, W4, W5, W6, W7};
    const float* Bs[7]  = {b1, b2, b3, b4, b5, b6, b7};

    // stage adjacency
    for (int i = t; i < 32 * APITCH; i += BLOCK) sA[i] = Apad_g[i];

    // load x tile: H(0,b,i) = x[b][i][0]; zero features 1..3 (layer-1 K padding)
    for (int idx = t; idx < BT * N_NODES; idx += BLOCK) {
        int b = idx / N_NODES, i = idx % N_NODES;
        hbuf[0 * HPITCH + b * APITCH + i] = x[(b0 + b) * N_NODES + i];
        hbuf[1 * HPITCH + b * APITCH + i] = 0.0f;
        hbuf[2 * HPITCH + b * APITCH + i] = 0.0f;
        hbuf[3 * HPITCH + b * APITCH + i] = 0.0f;
    }
    __syncthreads();

    // ---------------- 7 GCN layers (unrolled: dims become constants) --------
#pragma unroll
    for (int l = 0; l < 7; ++l) {
        const int fin = FIN[l], fout = FOUT[l];

        // stage weights (zero-padded 32x32) and bias
        for (int i = t; i < 32 * 32; i += BLOCK) sW[i] = 0.0f;
        if (t < 32) sBia[t] = (t < fout) ? Bs[l][t] : 0.0f;
        __syncthreads();
        for (int i = t; i < fin * fout; i += BLOCK) {
            int k = i / fout, n = i % fout;
            sW[k * 32 + n] = Ws[l][i];
        }
        __syncthreads();

        // ---- hw = h @ W : rows r = (b,node) -> 384 rows = 24 M-tiles ------
        const int ksteps = (fin + 3) >> 2;
        const int ntiles = (fout + 15) >> 4;
        // Incremental row->(b,i) tracking (row advances by 16 per M-tile).
        int ia = n16, ba = 0;            // A-frag row r0+n16 (r0=0: n16<24)
        int is = 8 * hi, bs = 0;         // D-store first row r0+8*hi
        for (int mt = 0; mt < 24; ++mt) {
            const int haddr = ba * APITCH + ia;       // lane stride 1: no conflicts
            for (int nt = 0; nt < ntiles; ++nt) {
                v8f acc = {};
                for (int ks = 0; ks < ksteps; ++ks) {
                    const int k0 = ks * 4 + 2 * hi;
                    v2f a, bf;
                    a[0]  = hbuf[(k0    ) * HPITCH + haddr];
                    a[1]  = hbuf[(k0 + 1) * HPITCH + haddr];
                    bf[0] = sW[(k0    ) * 32 + nt * 16 + n16];
                    bf[1] = sW[(k0 + 1) * 32 + nt * 16 + n16];
                    acc = wmma4(a, bf, acc);
                }
                const int n = nt * 16 + n16;          // lane stride WPITCH=401: coprime 64
                if (n < fout) {
                    float* wp = &wbuf[n * WPITCH + bs * APITCH + is];
                    int ii = is;
#pragma unroll
                    for (int v = 0; v < 8; ++v) {
                        *wp = acc[v];
                        // next row: +1 within a batch element, +2 across (23 -> pitch 25)
                        wp += (ii == N_NODES - 1) ? 2 : 1;
                        ii  = (ii == N_NODES - 1) ? 0 : ii + 1;
                    }
                }
            }
            // advance both row trackers by 16 (single conditional, no division)
            ia += 16; if (ia >= N_NODES) { ia -= N_NODES; ++ba; }
            is += 16; if (is >= N_NODES) { is -= N_NODES; ++bs; }
        }
        __syncthreads();

        // ---- h = relu(A @ hw + bias): per feature f, [M=16b, N=24i, K=24j] ----
        for (int f = 0; f < fout; ++f) {
            for (int nt = 0; nt < 2; ++nt) {
                v8f acc = {};
#pragma unroll
                for (int ks = 0; ks < 6; ++ks) {      // K = 24 = 6 * 4
                    const int j0 = ks * 4 + 2 * hi;
                    v2f a, bf;
                    a[0]  = wbuf[f * WPITCH + n16 * APITCH + j0];      // lane stride 25
                    a[1]  = wbuf[f * WPITCH + n16 * APITCH + j0 + 1];
                    bf[0] = sA[(nt * 16 + n16) * APITCH + j0];         // A^T[j][i]=A[i][j]
                    bf[1] = sA[(nt * 16 + n16) * APITCH + j0 + 1];
                    acc = wmma4(a, bf, acc);
                }
                const int i = nt * 16 + n16;          // lane stride 1: no conflicts
                if (i < N_NODES) {
                    const float bb = sBia[f];
                    float* hp = &hbuf[f * HPITCH + (8 * hi) * APITCH + i];
#pragma unroll
                    for (int v = 0; v < 8; ++v) {
                        *hp = fmaxf(acc[v] + bb, 0.0f);
                        hp += APITCH;                 // next batch row
                    }
                }
            }
        }
        __syncthreads();
    }

    // ---------------- fc1: [16,768] @ [768,128], K = 192 * 4 ----------------
    // h768[b][k] with k = i*32+f lives at H(k&31, b, k>>5); k0 ≡ 0,2 (mod 4)
    // so k0 and k0+1 share the same node index (k>>5).
    v8f facc[8] = {};
    for (int ks = 0; ks < 192; ++ks) {
        const int k0 = ks * 4 + 2 * hi;
        const int ia = k0 >> 5, fa = k0 & 31;
        v2f a;
        a[0] = hbuf[(fa    ) * HPITCH + n16 * APITCH + ia];  // lane stride 25
        a[1] = hbuf[(fa + 1) * HPITCH + n16 * APITCH + ia];
#pragma unroll
        for (int nt = 0; nt < 8; ++nt) {
            v2f bf;
            bf[0] = fcw1[(long)(k0    ) * 128 + nt * 16 + n16];
            bf[1] = fcw1[(long)(k0 + 1) * 128 + nt * 16 + n16];
            facc[nt] = wmma4(a, bf, facc[nt]);
        }
    }
    __syncthreads();   // wbuf is dead; sFc overlays it
#pragma unroll
    for (int nt = 0; nt < 8; ++nt) {
        const int n = nt * 16 + n16;
        const float bb = fcb1[n];
#pragma unroll
        for (int v = 0; v < 8; ++v)
            sFc[(v + 8 * hi) * FCPITCH + n] = facc[nt][v] + bb;
    }
    __syncthreads();

    // ---------------- fc2 (128 -> 2) + log_softmax, VALU ----------------
    {
        const int b = n16;     // batch element in tile
        const int c = hi;      // class 0/1
        float z = fcb2[c];
        for (int k = 0; k < 128; ++k)
            z += sFc[b * FCPITCH + k] * fcw2[k * 2 + c];
        const float other = __shfl_xor(z, 16);
        const float mx  = fmaxf(z, other);
        const float lse = mx + logf(expf(z - mx) + expf(other - mx));
        out[(b0 + b) * 2 + c] = z - lse;
    }
}

extern "C" void kernel_launch(void* const* d_in, const int* in_sizes, int n_in,
                              void* d_out, int out_size, void* d_ws, size_t ws_size,
                              hipStream_t stream) {
    const float* x    = (const float*)d_in[0];
    const int*   ei   = (const int*)d_in[1];
    const float* W[7]; const float* B[7];
    for (int i = 0; i < 7; ++i) {
        W[i] = (const float*)d_in[2 + 2 * i];
        B[i] = (const float*)d_in[3 + 2 * i];
    }
    const float* fcw1 = (const float*)d_in[16];
    const float* fcb1 = (const float*)d_in[17];
    const float* fcw2 = (const float*)d_in[18];
    const float* fcb2 = (const float*)d_in[19];
    float* out = (float*)d_out;
    float* Apad = (float*)d_ws;  // 32*25 floats

    build_adj_kernel<<<1, 128, 0, stream>>>(ei, Apad);

    const size_t smem = 27488 * sizeof(float);  // ~110 KB dynamic LDS (<= 320 KB WGP LDS)
    (void)hipFuncSetAttribute((const void*)gcn_fused_kernel,
                              hipFuncAttributeMaxDynamicSharedMemorySize, (int)smem);

    const int nblocks = 32768 / BT;  // 2048
    gcn_fused_kernel<<<nblocks, BLOCK, smem, stream>>>(
        x, Apad,
        W[0], B[0], W[1], B[1], W[2], B[2], W[3], B[3],
        W[4], B[4], W[5], B[5], W[6], B[6],
        fcw1, fcb1, fcw2, fcb2, out);
}